// curve2D_41248865911228
// MI455X (gfx1250) — compile-verified
//
#include <hip/hip_runtime.h>
#include <hip/hip_bf16.h>
#include <math.h>

// ---------------------------------------------------------------------------
// Problem constants (from the reference)
// ---------------------------------------------------------------------------
#define NB   256      // batches
#define NU   1024     // data points per batch
#define NT   1024     // curve points per batch (8 segments x 128)
#define ALPHA_C   0.5f
#define LAM_LEN_C 1e-4f
#define LAM_MOM_C 1e-3f
#define EPS_C     1e-8f
#define IMG_INV   (1.0f / 28.0f)

typedef float v2f __attribute__((ext_vector_type(2)));
typedef float v8f __attribute__((ext_vector_type(8)));

__device__ __forceinline__ unsigned f2key(float f) {
    unsigned b = __float_as_uint(f);
    return (b & 0x80000000u) ? ~b : (b | 0x80000000u);
}
__device__ __forceinline__ float key2f(unsigned k) {
    unsigned b = (k & 0x80000000u) ? (k ^ 0x80000000u) : ~k;
    return __uint_as_float(b);
}

// Balanced min-tree over a 16x16 WMMA D tile (8 regs): depth 3.
__device__ __forceinline__ float tile_min(const v8f& d) {
    float m0 = fminf(d[0], d[1]);
    float m1 = fminf(d[2], d[3]);
    float m2 = fminf(d[4], d[5]);
    float m3 = fminf(d[6], d[7]);
    m0 = fminf(m0, m1);
    m2 = fminf(m2, m3);
    return fminf(m0, m2);
}

// One block per batch. 256 threads = 8 waves. Each wave owns 8 u-tiles of 16,
// loops over 64 t-tiles of 16, one V_WMMA_F32_16X16X4_F32 per (u-tile,t-tile).
// dist(u,t) = [-2x_u, -2y_u, |u|^2, 1] . [x_t, y_t, 1, |t|^2]  (K = 4 exact fit)
__global__ __launch_bounds__(256) void curve2D_loss_kernel(
    const float* __restrict__ P,     // [B,U,2]  pixel coords
    const float* __restrict__ W,     // [B,U]    weights
    const float* __restrict__ outs,  // [B,36]   spline params
    float* __restrict__ part)        // [5*B] partials: s1, rev, len, mom, wsum
{
    __shared__ float    tx[NT], ty[NT];
    __shared__ v2f      bp[2 * NT];   // prepacked B operands: [t]=(x,y), [NT+t]=(1,|t|^2)
    __shared__ unsigned mrk[NT];      // per-t min key (order-preserving uint)
    __shared__ float    locx[9], locy[9], momx[9], momy[9];
    __shared__ float    red[5][8];

    const int b    = blockIdx.x;
    const int tid  = threadIdx.x;
    const int lane = tid & 31;
    const int wav  = tid >> 5;
    const int lo   = lane & 15;
    const bool hi  = lane >= 16;

    const float* ob = outs + b * 36;

    // ---- decode control points + Hermite moments (9 nodes) ----
    if (tid < 9) {
        const int s = tid;
        locx[s] = ob[2 * s];
        locy[s] = ob[2 * s + 1];
        const int km = (s == 0) ? 0 : (s - 1);
        const int kp = (s == 8) ? 7 : s;
        const float dlpx = ob[2 * (km + 1)]     - ob[2 * km];
        const float dlpy = ob[2 * (km + 1) + 1] - ob[2 * km + 1];
        const float dlnx = ob[2 * (kp + 1)]     - ob[2 * kp];
        const float dlny = ob[2 * (kp + 1) + 1] - ob[2 * kp + 1];
        momx[s] = ob[18 + 2 * s]     + 0.5f * (dlpx + dlnx);
        momy[s] = ob[18 + 2 * s + 1] + 0.5f * (dlpy + dlny);
    }
    __syncthreads();

    // ---- evaluate curve points into LDS; prepack WMMA B operands ----
    for (int t = tid; t < NT; t += 256) {
        const int s = t >> 7, j = t & 127;
        const float tp  = (float)(j * (1.0 / 127.0));
        const float tp2 = tp * tp, tp3 = tp2 * tp;
        float p0 = locx[s], p1 = locx[s + 1], m0 = momx[s], m1 = momx[s + 1];
        const float cx = p0 + m0 * tp
                       + (-3.f * p0 + 3.f * p1 - 2.f * m0 - m1) * tp2
                       + ( 2.f * p0 - 2.f * p1 +      m0 + m1) * tp3;
        p0 = locy[s]; p1 = locy[s + 1]; m0 = momy[s]; m1 = momy[s + 1];
        const float cy = p0 + m0 * tp
                       + (-3.f * p0 + 3.f * p1 - 2.f * m0 - m1) * tp2
                       + ( 2.f * p0 - 2.f * p1 +      m0 + m1) * tp3;
        tx[t] = cx;
        ty[t] = cy;
        bp[t]      = (v2f){cx, cy};                   // B rows K=0,1 (lanes 0-15)
        bp[NT + t] = (v2f){1.0f, cx * cx + cy * cy};  // B rows K=2,3 (lanes 16-31)
        mrk[t] = 0xFFFFFFFFu;
    }
    __syncthreads();

    // ---- cheap regularizer terms + W partial sum ----
    float acc_len = 0.f, acc_mom = 0.f, acc_w = 0.f;
    for (int t = tid; t < NT - 1; t += 256) {
        const float dx = tx[t] - tx[t + 1], dy = ty[t] - ty[t + 1];
        float d2 = dx * dx + dy * dy;
        d2 = fminf(fmaxf(d2, EPS_C), 1.0f / EPS_C);
        acc_len += sqrtf(d2);
    }
    if (tid < 18) { const float dm = ob[18 + tid]; acc_mom = dm * dm; }
    for (int u = tid; u < NU; u += 256) acc_w += W[b * NU + u];

    // ---- register-resident A operands (8 u-tiles per wave) ----
    // A (16x4 f32): lanes 0-15 -> K=0,1 ; lanes 16-31 -> K=2,3
    v2f a[8];
    v8f umin[8];
    const float INF = __builtin_inff();
    #pragma unroll
    for (int i = 0; i < 8; i++) {
        const int u = (wav * 8 + i) * 16 + lo;
        const float ux = P[(b * NU + u) * 2 + 0] * IMG_INV;
        const float uy = P[(b * NU + u) * 2 + 1] * IMG_INV;
        const float u2 = ux * ux + uy * uy;
        a[i].x = hi ? u2   : -2.0f * ux;
        a[i].y = hi ? 1.0f : -2.0f * uy;
        umin[i] = (v8f){INF, INF, INF, INF, INF, INF, INF, INF};
    }

    const v8f czero = (v8f){0.f, 0.f, 0.f, 0.f, 0.f, 0.f, 0.f, 0.f};
    const int bofs = hi ? NT : 0;   // lane-half selects the prepacked B row pair

    // Software-pipelined B operand: prefetch tile tt+1 before the WMMA block of
    // tile tt; the ds_bpermute's dscnt wait (after the WMMAs) retires it for free.
    v2f bb = bp[bofs + lo];
    for (int tt = 0; tt < 64; tt++) {
        const v2f bbn = bp[bofs + (((tt + 1) & 63) * 16 + lo)];
        float tmin[8];
        #pragma unroll
        for (int i = 0; i < 8; i++) {
            v8f d = __builtin_amdgcn_wmma_f32_16x16x4_f32(
                false, a[i], false, bb, (short)0, czero, false, false);
            #pragma unroll
            for (int r = 0; r < 8; r++) umin[i][r] = fminf(umin[i][r], d[r]);
            tmin[i] = tile_min(d);
        }
        // balanced tree over the 8 u-tiles (depth 3), then fold M-halves
        float c0 = fminf(tmin[0], tmin[1]);
        float c1 = fminf(tmin[2], tmin[3]);
        float c2 = fminf(tmin[4], tmin[5]);
        float c3 = fminf(tmin[6], tmin[7]);
        c0 = fminf(c0, c1);
        c2 = fminf(c2, c3);
        float cm = fminf(c0, c2);
        cm = fminf(cm, __shfl_xor(cm, 16, 32));
        if (!hi) atomicMin(&mrk[tt * 16 + lo], f2key(cm));
        bb = bbn;
    }

    // ---- per-u min over all t: reduce across the 16-lane halves ----
    float s1 = 0.f;
    #pragma unroll
    for (int i = 0; i < 8; i++) {
        #pragma unroll
        for (int r = 0; r < 8; r++) {
            float v = umin[i][r];
            v = fminf(v, __shfl_xor(v, 1, 32));
            v = fminf(v, __shfl_xor(v, 2, 32));
            v = fminf(v, __shfl_xor(v, 4, 32));
            v = fminf(v, __shfl_xor(v, 8, 32));
            if (lo == 0) {
                const int u = (wav * 8 + i) * 16 + r + (hi ? 8 : 0);
                s1 += W[b * NU + u] * v;
            }
        }
    }

    __syncthreads();
    float acc_rev = 0.f;
    for (int t = tid; t < NT; t += 256) acc_rev += key2f(mrk[t]);

    // ---- deterministic block reduction of the 5 partials ----
    float vals[5] = {s1, acc_rev, acc_len, acc_mom, acc_w};
    #pragma unroll
    for (int k = 0; k < 5; k++) {
        float v = vals[k];
        #pragma unroll
        for (int off = 16; off > 0; off >>= 1) v += __shfl_xor(v, off, 32);
        if (lane == 0) red[k][wav] = v;
    }
    __syncthreads();
    if (tid < 5) {
        float v = 0.f;
        #pragma unroll
        for (int wv = 0; wv < 8; wv++) v += red[tid][wv];
        part[tid * NB + b] = v;
    }
}

__global__ __launch_bounds__(256) void curve2D_finalize_kernel(
    const float* __restrict__ part, float* __restrict__ out)
{
    __shared__ float red[5][8];
    const int tid = threadIdx.x, lane = tid & 31, wav = tid >> 5;
    #pragma unroll
    for (int k = 0; k < 5; k++) {
        float x = part[k * NB + tid];
        #pragma unroll
        for (int off = 16; off > 0; off >>= 1) x += __shfl_xor(x, off, 32);
        if (lane == 0) red[k][wav] = x;
    }
    __syncthreads();
    if (tid == 0) {
        float tot[5];
        #pragma unroll
        for (int k = 0; k < 5; k++) {
            float s = 0.f;
            #pragma unroll
            for (int w = 0; w < 8; w++) s += red[k][w];
            tot[k] = s;
        }
        const float invB = 1.0f / (float)NB;
        float loss = tot[0] / ((float)NB * tot[4])   // weighted Chamfer (global W norm)
                   + ALPHA_C   * tot[1] * invB       // reverse Chamfer
                   + LAM_LEN_C * tot[2] * invB       // curve length
                   + LAM_MOM_C * tot[3] * invB;      // momentum penalty
        out[0] = loss;
    }
}

extern "C" void kernel_launch(void* const* d_in, const int* in_sizes, int n_in,
                              void* d_out, int out_size, void* d_ws, size_t ws_size,
                              hipStream_t stream) {
    const float* P    = (const float*)d_in[0];   // [B,U,2] f32
    const float* W    = (const float*)d_in[1];   // [B,U]   f32
    const float* outs = (const float*)d_in[2];   // [B,36]  f32
    float* part = (float*)d_ws;                  // 5*B floats, overwritten each call

    curve2D_loss_kernel<<<NB, 256, 0, stream>>>(P, W, outs, part);
    curve2D_finalize_kernel<<<1, 256, 0, stream>>>(part, (float*)d_out);
}